// GCNII_17463337026196
// MI455X (gfx1250) — compile-verified
//
#include <hip/hip_runtime.h>
#include <hip/hip_bf16.h>
#include <math.h>

#define N_NODES 100000
#define E_EDGES 3200000
#define NIN     500
#define NHID    64
#define NOUT    40
#define NLAYERS 8
#define ALPHA   0.1f

// K-pair-major LDS stride (floats) for B fragments:
// pair (k,k+1) row lives at Wp[(k/2)*BRS + 2n + (k&1)].
// BRS = 160 => half-wave regions 640B apart => disjoint LDS bank halves.
#define BRS 160

typedef __attribute__((ext_vector_type(2))) float v2f;
typedef __attribute__((ext_vector_type(8))) float v8f;

// V_WMMA_F32_16X16X4_F32 with D tied to C ("+v") so the register allocator
// cannot split the accumulator across two tuples (kills the per-iteration
// v_mov_b64 rotation the pipeliner generated around the intrinsic form).
// D->C chaining is the architected accumulate path (no NOPs needed; hazard
// table only covers D->A/B and D->VALU reads).
__device__ __forceinline__ void wmma_acc(v8f& c, v2f a, v2f b) {
  asm("v_wmma_f32_16x16x4_f32 %0, %1, %2, %0" : "+v"(c) : "v"(a), "v"(b));
}
// Worst-case WMMA->VALU RAW guard (9 wait states) before epilogue reads D.
__device__ __forceinline__ void wmma_epilogue_guard() {
  asm volatile("s_nop 8");
}

// ---------------------------------------------------------------------------
// Kernel 1: h0 = relu(x @ W_in + b_in)   [N,500]x[500,64]
// One wave computes a 32x64 output tile (2 A-frags x 4 col-tiles = 8 WMMA
// accumulators), K-loop step 4. W_in pre-swizzled into 160 KB dynamic LDS so
// each B fragment is a single aligned ds_load_b64 (no VGPR packing moves).
// ---------------------------------------------------------------------------
__global__ void __launch_bounds__(256) in_gemm_kernel(
    const float* __restrict__ x, const float* __restrict__ Win,
    const float* __restrict__ bin, float* __restrict__ h0) {
  extern __shared__ float Wp[];                 // (NIN/2)*BRS floats
  for (int idx = threadIdx.x; idx < NIN * NHID; idx += 256) {
    const int k = idx >> 6;                     // global row-major [500,64]
    const int n = idx & 63;
    Wp[(k >> 1) * BRS + 2 * n + (k & 1)] = Win[idx];
  }
  __syncthreads();

  const int wave = threadIdx.x >> 5;
  const int lane = threadIdx.x & 31;
  const int half = lane >> 4;                   // 0: K=k,k+1  1: K=k+2,k+3
  const int l16  = lane & 15;
  const int r0   = blockIdx.x * 256 + wave * 32;

  const int arow0 = min(r0 + l16,      N_NODES - 1);  // A row (M = lane%16)
  const int arow1 = min(r0 + 16 + l16, N_NODES - 1);
  const float* xr0 = x + (long long)arow0 * NIN;
  const float* xr1 = x + (long long)arow1 * NIN;

  v8f acc0[4] = {};
  v8f acc1[4] = {};
  for (int k = 0; k < NIN; k += 4) {
    const int kk = k + 2 * half;
    const v2f a0 = *reinterpret_cast<const v2f*>(xr0 + kk);   // global_load_b64
    const v2f a1 = *reinterpret_cast<const v2f*>(xr1 + kk);
    const int bbase = (kk >> 1) * BRS + 2 * l16;
#pragma unroll
    for (int t = 0; t < 4; ++t) {
      const v2f b = *reinterpret_cast<const v2f*>(&Wp[bbase + t * 32]); // ds_load_b64
      wmma_acc(acc0[t], a0, b);
      wmma_acc(acc1[t], a1, b);
    }
  }
  wmma_epilogue_guard();

  // epilogue: bias + ReLU. C layout: row = vgpr + 8*half, col = lane%16
#pragma unroll
  for (int t = 0; t < 4; ++t) {
    const int n = t * 16 + l16;
    const float bias = bin[n];
#pragma unroll
    for (int r = 0; r < 8; ++r) {
      const int m = r + 8 * half;
      const int orow0 = r0 + m;
      const int orow1 = r0 + 16 + m;
      if (orow0 < N_NODES) {
        const float v = acc0[t][r] + bias;
        h0[(long long)orow0 * NHID + n] = v > 0.f ? v : 0.f;
      }
      if (orow1 < N_NODES) {
        const float v = acc1[t][r] + bias;
        h0[(long long)orow1 * NHID + n] = v > 0.f ? v : 0.f;
      }
    }
  }
}

// ---------------------------------------------------------------------------
// Zero-fill (hB must be zero before each scatter-add SpMM)
// ---------------------------------------------------------------------------
__global__ void zero_kernel(float4* __restrict__ p, int n4) {
  int i = blockIdx.x * blockDim.x + threadIdx.x;
  if (i < n4) p[i] = make_float4(0.f, 0.f, 0.f, 0.f);
}

// ---------------------------------------------------------------------------
// Kernel 2: SpMM  hi[row] += edge_weight * h[col]   (COO, 64 channels/edge)
// 16 threads per edge, float4 gathers (h is L2-resident: 25.6 MB < 192 MB),
// f32 HW atomics scatter (no-return -> STOREcnt path).
// ---------------------------------------------------------------------------
__global__ void __launch_bounds__(256) spmm_kernel(
    const int* __restrict__ row, const int* __restrict__ col,
    const float* __restrict__ ew, const float* __restrict__ h,
    float* __restrict__ hi) {
  const long long t = (long long)blockIdx.x * blockDim.x + threadIdx.x;
  const long long total = (long long)E_EDGES * 16;
  if (t >= total) return;
  const int e = (int)(t >> 4);
  const int q = (int)(t & 15);
  const int c = col[e];
  const int r = row[e];
  const float w = ew[e];
  const float4 v = *reinterpret_cast<const float4*>(h + (long long)c * NHID + q * 4);
  float* dst = hi + (long long)r * NHID + q * 4;
  unsafeAtomicAdd(dst + 0, v.x * w);   // global_atomic_add_f32
  unsafeAtomicAdd(dst + 1, v.y * w);
  unsafeAtomicAdd(dst + 2, v.z * w);
  unsafeAtomicAdd(dst + 3, v.w * w);
}

// ---------------------------------------------------------------------------
// Kernel 3: GCNII layer body.
//   S = 0.9*hi + 0.1*h0
//   h = relu(beta*(S @ Wl) + (1-beta)*S)
// Residual folded into accumulator init: acc = ((1-beta)/beta)*S, then
// D = S@Wl + acc, h = relu(beta*D). 32 rows/wave, pre-swizzled B in LDS.
// ---------------------------------------------------------------------------
__global__ void __launch_bounds__(256) conv_kernel(
    const float* __restrict__ hi, const float* __restrict__ h0,
    const float* __restrict__ Wl, float beta, float* __restrict__ hout) {
  __shared__ float Wp[(NHID / 2) * BRS];        // 32*160*4 = 20.5 KB
  for (int idx = threadIdx.x; idx < NHID * NHID; idx += 256) {
    const int k = idx >> 6;
    const int n = idx & 63;
    Wp[(k >> 1) * BRS + 2 * n + (k & 1)] = Wl[idx];
  }
  __syncthreads();

  const int wave = threadIdx.x >> 5;
  const int lane = threadIdx.x & 31;
  const int half = lane >> 4;
  const int l16  = lane & 15;
  const int r0   = blockIdx.x * 256 + wave * 32;

  const float sA = 1.f - ALPHA;                 // 0.9
  const float sB = ALPHA;                       // 0.1
  const float cscale = (1.f - beta) / beta;

  // accumulator init with residual term, in C layout
  v8f acc0[4] = {};
  v8f acc1[4] = {};
#pragma unroll
  for (int t = 0; t < 4; ++t) {
    const int n = t * 16 + l16;
#pragma unroll
    for (int r = 0; r < 8; ++r) {
      const int m = r + 8 * half;
      const long long rr0 = min(r0 + m,      N_NODES - 1);
      const long long rr1 = min(r0 + 16 + m, N_NODES - 1);
      acc0[t][r] = cscale * (sA * hi[rr0 * NHID + n] + sB * h0[rr0 * NHID + n]);
      acc1[t][r] = cscale * (sA * hi[rr1 * NHID + n] + sB * h0[rr1 * NHID + n]);
    }
  }

  const long long arow0 = min(r0 + l16,      N_NODES - 1);
  const long long arow1 = min(r0 + 16 + l16, N_NODES - 1);
  const float* hia0 = hi + arow0 * NHID;
  const float* h0a0 = h0 + arow0 * NHID;
  const float* hia1 = hi + arow1 * NHID;
  const float* h0a1 = h0 + arow1 * NHID;
#pragma unroll
  for (int k = 0; k < NHID; k += 4) {
    const int kk = k + 2 * half;
    const v2f hv0 = *reinterpret_cast<const v2f*>(hia0 + kk);
    const v2f gv0 = *reinterpret_cast<const v2f*>(h0a0 + kk);
    const v2f hv1 = *reinterpret_cast<const v2f*>(hia1 + kk);
    const v2f gv1 = *reinterpret_cast<const v2f*>(h0a1 + kk);
    v2f a0, a1;
    a0.x = sA * hv0.x + sB * gv0.x;  a0.y = sA * hv0.y + sB * gv0.y;
    a1.x = sA * hv1.x + sB * gv1.x;  a1.y = sA * hv1.y + sB * gv1.y;
    const int bbase = (kk >> 1) * BRS + 2 * l16;
#pragma unroll
    for (int t = 0; t < 4; ++t) {
      const v2f b = *reinterpret_cast<const v2f*>(&Wp[bbase + t * 32]);
      wmma_acc(acc0[t], a0, b);
      wmma_acc(acc1[t], a1, b);
    }
  }
  wmma_epilogue_guard();

#pragma unroll
  for (int t = 0; t < 4; ++t) {
    const int n = t * 16 + l16;
#pragma unroll
    for (int r = 0; r < 8; ++r) {
      const int m = r + 8 * half;
      const int orow0 = r0 + m;
      const int orow1 = r0 + 16 + m;
      if (orow0 < N_NODES) {
        const float v = beta * acc0[t][r];
        hout[(long long)orow0 * NHID + n] = v > 0.f ? v : 0.f;
      }
      if (orow1 < N_NODES) {
        const float v = beta * acc1[t][r];
        hout[(long long)orow1 * NHID + n] = v > 0.f ? v : 0.f;
      }
    }
  }
}

// ---------------------------------------------------------------------------
// Kernel 4: logits = h @ W_out + b_out; out = log_softmax(logits)
// One thread per node row; W_out (10 KB) staged in LDS (broadcast reads).
// ---------------------------------------------------------------------------
__global__ void __launch_bounds__(256) out_kernel(
    const float* __restrict__ h, const float* __restrict__ Wout,
    const float* __restrict__ bout, float* __restrict__ out) {
  __shared__ float Ws[NHID * NOUT];             // 2560 floats = 10 KB
  __shared__ float bs[NOUT];
  for (int idx = threadIdx.x; idx < NHID * NOUT; idx += 256) Ws[idx] = Wout[idx];
  if (threadIdx.x < NOUT) bs[threadIdx.x] = bout[threadIdx.x];
  __syncthreads();

  const int i = blockIdx.x * blockDim.x + threadIdx.x;
  if (i >= N_NODES) return;

  float hr[NHID];
  const float4* hp = reinterpret_cast<const float4*>(h + (long long)i * NHID);
#pragma unroll
  for (int k4 = 0; k4 < NHID / 4; ++k4) {
    const float4 v = hp[k4];
    hr[4 * k4 + 0] = v.x; hr[4 * k4 + 1] = v.y;
    hr[4 * k4 + 2] = v.z; hr[4 * k4 + 3] = v.w;
  }

  float lg[NOUT];
#pragma unroll
  for (int c = 0; c < NOUT; ++c) lg[c] = bs[c];
  for (int k = 0; k < NHID; ++k) {
    const float hk = hr[k];
#pragma unroll
    for (int c = 0; c < NOUT; ++c) lg[c] = fmaf(hk, Ws[k * NOUT + c], lg[c]);
  }

  float mx = lg[0];
#pragma unroll
  for (int c = 1; c < NOUT; ++c) mx = fmaxf(mx, lg[c]);
  float s = 0.f;
#pragma unroll
  for (int c = 0; c < NOUT; ++c) s += expf(lg[c] - mx);
  const float lse = mx + logf(s);
#pragma unroll
  for (int c = 0; c < NOUT; ++c)
    out[(long long)i * NOUT + c] = lg[c] - lse;
}

// ---------------------------------------------------------------------------
extern "C" void kernel_launch(void* const* d_in, const int* in_sizes, int n_in,
                              void* d_out, int out_size, void* d_ws, size_t ws_size,
                              hipStream_t stream) {
  (void)in_sizes; (void)n_in; (void)out_size; (void)ws_size;
  const float* x     = (const float*)d_in[0];
  const int*   row   = (const int*)  d_in[1];
  const int*   col   = (const int*)  d_in[2];
  const float* ew    = (const float*)d_in[3];
  const float* Win   = (const float*)d_in[4];
  const float* bin   = (const float*)d_in[5];
  const float* convW = (const float*)d_in[6];
  const float* Wout  = (const float*)d_in[7];
  const float* bout  = (const float*)d_in[8];

  const size_t S = (size_t)N_NODES * NHID;      // elements per feature buffer
  float* h0 = (float*)d_ws;
  float* hA = h0 + S;
  float* hB = hA + S;

  const int rowBlocks = (N_NODES + 255) / 256;  // 391 (32 rows/wave, 8 waves)
  const size_t inLds = (size_t)(NIN / 2) * BRS * sizeof(float);   // 160 KB
  in_gemm_kernel<<<rowBlocks, 256, inLds, stream>>>(x, Win, bin, h0);

  const long long spmmItems = (long long)E_EDGES * 16;
  const int spmmBlocks = (int)((spmmItems + 255) / 256);
  const int zeroBlocks = (int)((S / 4 + 255) / 256);

  for (int l = 0; l < NLAYERS; ++l) {
    const float beta = logf(0.5f / (float)(l + 1) + 1.f);
    zero_kernel<<<zeroBlocks, 256, 0, stream>>>((float4*)hB, (int)(S / 4));
    spmm_kernel<<<spmmBlocks, 256, 0, stream>>>(row, col, ew, (l == 0) ? h0 : hA, hB);
    conv_kernel<<<rowBlocks, 256, 0, stream>>>(hB, h0, convW + (size_t)l * NHID * NHID,
                                               beta, hA);
  }

  out_kernel<<<(N_NODES + 255) / 256, 256, 0, stream>>>(hA, Wout, bout, (float*)d_out);
}